// GCNConv_dgl_1932735283961
// MI455X (gfx1250) — compile-verified
//
#include <hip/hip_runtime.h>

#define N_NODES 100000
#define N_EDGES 1600000
#define D_IN    128
#define D_OUT   64

typedef __attribute__((ext_vector_type(2))) float v2f;
typedef __attribute__((ext_vector_type(8))) float v8f;

// ---------------------------------------------------------------------------
// Zero-fill the output (it is poisoned with 0xAA; scatter-add needs zeros).
// ---------------------------------------------------------------------------
__global__ __launch_bounds__(256) void zero_kernel(float4* __restrict__ out, int n4) {
    int i = blockIdx.x * blockDim.x + threadIdx.x;
    if (i < n4) out[i] = make_float4(0.f, 0.f, 0.f, 0.f);
}

// ---------------------------------------------------------------------------
// h = x @ W + b   via V_WMMA_F32_16X16X4_F32 (full fp32, matches reference).
//
// Register-blocked over N: each wave computes a 16x64 strip of h (4 C-tiles,
// 32 accumulator VGPRs), so each x row is loaded exactly once (51.2 MB total
// A-traffic = the HBM roofline floor) and one A fragment feeds 4 WMMAs per
// K-step. W (32 KB) is the only re-read operand and lives in WGP$/L2.
//
// ISA layouts (CDNA5 ISA §7.12.2):
//   A 16x4 f32 : lanes 0-15 -> M=lane, VGPR0=K0, VGPR1=K1
//                lanes 16-31 -> M=lane-16, VGPR0=K2, VGPR1=K3
//   B 4x16 f32 : row K striped across lanes within a VGPR (mirror of A)
//   C/D 16x16  : VGPR v, lanes 0-15 -> M=v,  N=lane
//                         lanes 16-31 -> M=v+8, N=lane-16
// ---------------------------------------------------------------------------
__global__ __launch_bounds__(256) void gemm_bias_kernel(
    const float* __restrict__ x, const float* __restrict__ W,
    const float* __restrict__ bias, float* __restrict__ h)
{
    const int lane = threadIdx.x & 31;
    const int wave = threadIdx.x >> 5;
    const int m0   = (blockIdx.x * 8 + wave) * 16;     // 16-row strip per wave
    if (m0 >= N_NODES) return;   // wave-uniform: EXEC stays all-1s for WMMA

    const int half = lane >> 4;         // 0 for lanes 0-15, 1 for 16-31
    const int l15  = lane & 15;

    // Seed accumulators with bias: D[m, n] starts at b[n] -> bias add is free.
    v8f acc[4];
    #pragma unroll
    for (int t = 0; t < 4; ++t) {
        const float bval = bias[t * 16 + l15];
        #pragma unroll
        for (int i = 0; i < 8; ++i) acc[t][i] = bval;
    }

    // Per-lane base pointers per the A/B VGPR layouts above.
    const float* xrow  = x + (size_t)(m0 + l15) * D_IN + half * 2;
    const float* wbase = W + (size_t)(half * 2) * D_OUT + l15;

    #pragma unroll
    for (int k0 = 0; k0 < D_IN; k0 += 4) {
        v2f a;
        a.x = xrow[k0 + 0];                 // K = k0 + 2*half
        a.y = xrow[k0 + 1];                 // K = k0 + 2*half + 1
        #pragma unroll
        for (int t = 0; t < 4; ++t) {       // 4 N-tiles share one A fragment
            v2f bm;
            bm.x = wbase[(size_t)(k0 + 0) * D_OUT + t * 16];
            bm.y = wbase[(size_t)(k0 + 1) * D_OUT + t * 16];
            acc[t] = __builtin_amdgcn_wmma_f32_16x16x4_f32(
                         /*neg_a=*/false, a, /*neg_b=*/false, bm,
                         /*c_mod=*/(short)0, acc[t],
                         /*reuse_a=*/false, /*reuse_b=*/false);
        }
    }

    // Store the 16x64 strip (bias already folded in).
    float* hout = h + (size_t)(m0 + half * 8) * D_OUT + l15;
    #pragma unroll
    for (int t = 0; t < 4; ++t)
        #pragma unroll
        for (int v = 0; v < 8; ++v)
            hout[(size_t)v * D_OUT + t * 16] = acc[t][v];
}

// ---------------------------------------------------------------------------
// Edge phase: one wave32 per edge. Lane l handles elements 2l, 2l+1 of the
// 64-float source row (one coalesced 256B row read, L2-resident), scales by
// edge_w, and scatter-adds with GLOBAL_ATOMIC_ADD_F32 (resolved in L2:
// h + out = 51.2 MB << 192 MB global L2).
// ---------------------------------------------------------------------------
__global__ __launch_bounds__(256) void scatter_kernel(
    const float* __restrict__ h, const float* __restrict__ edge_w,
    const int* __restrict__ src, const int* __restrict__ dst,
    float* __restrict__ out)
{
    const int lane = threadIdx.x & 31;
    const int e    = blockIdx.x * 8 + (threadIdx.x >> 5);
    if (e >= N_EDGES) return;

    const int   s = src[e];   // uniform across the wave -> single L1 request
    const int   d = dst[e];
    const float w = edge_w[e];

    const float2 v = ((const float2*)(h + (size_t)s * D_OUT))[lane];
    float* orow = out + (size_t)d * D_OUT + lane * 2;
    atomicAdd(orow + 0, v.x * w);
    atomicAdd(orow + 1, v.y * w);
}

// ---------------------------------------------------------------------------
extern "C" void kernel_launch(void* const* d_in, const int* in_sizes, int n_in,
                              void* d_out, int out_size, void* d_ws, size_t ws_size,
                              hipStream_t stream) {
    const float* x      = (const float*)d_in[0];
    const float* W      = (const float*)d_in[1];
    const float* b      = (const float*)d_in[2];
    const float* edge_w = (const float*)d_in[3];
    const int*   src    = (const int*)d_in[4];
    const int*   dst    = (const int*)d_in[5];
    float*       out    = (float*)d_out;
    float*       h      = (float*)d_ws;   // 100000*64*4 = 25.6 MB scratch

    (void)in_sizes; (void)n_in; (void)ws_size;

    // 1) zero the output accumulator
    const int n4 = out_size / 4;                     // 6.4M floats -> 1.6M float4
    zero_kernel<<<(n4 + 255) / 256, 256, 0, stream>>>((float4*)out, n4);

    // 2) h = x @ W + b  (WMMA fp32, one 16x64 strip per wave)
    const int m_tiles = (N_NODES + 15) / 16;               // 6250
    const int gemm_blocks = (m_tiles + 7) / 8;             // 782
    gemm_bias_kernel<<<gemm_blocks, 256, 0, stream>>>(x, W, b, h);

    // 3) gather * edge_w, scatter-add into out
    scatter_kernel<<<(N_EDGES + 7) / 8, 256, 0, stream>>>(h, edge_w, src, dst, out);
}